// GATModel_75428215652386
// MI455X (gfx1250) — compile-verified
//
#include <hip/hip_runtime.h>

typedef __attribute__((ext_vector_type(2))) float v2f;
typedef __attribute__((ext_vector_type(8))) float v8f;

#define IN_CH 128
#define OUT_CH 64
#define NEG_SLOPE 0.2f
#define N_GRAPHS 64
#define POOL_CHUNK 2048

// ---------------------------------------------------------------------------
// Kernel 1: h = x @ W via V_WMMA_F32_16X16X4_F32, fused a_src/a_dst projections.
// One wave per 16-row tile; 4 C-tiles cover the 64 output channels.
// A (16x4 f32):  lane m = lane&15 holds row m; VGPR j holds K = k0 + j + 2*(lane>>4)
// B (4x16 f32):  lane m holds col m; VGPR j holds K = k0 + j + 2*(lane>>4)
// C/D (16x16):   VGPR v holds row v + 8*(lane>>4), col = lane&15
// ---------------------------------------------------------------------------
__global__ __launch_bounds__(128) void gat_gemm_wmma(
    const float* __restrict__ x, const float* __restrict__ W,
    const float* __restrict__ att_src, const float* __restrict__ att_dst,
    float* __restrict__ h, float* __restrict__ asrc, float* __restrict__ adst,
    int nrows)
{
    const int wave = threadIdx.x >> 5;
    const int lane = threadIdx.x & 31;
    const int row_base = (blockIdx.x * 4 + wave) * 16;
    if (row_base >= nrows) return;   // wave-uniform: EXEC stays all-ones for WMMA
    const int m = lane & 15;
    const int half = lane >> 4;

    v8f acc[4] = {v8f{0}, v8f{0}, v8f{0}, v8f{0}};
    const float* xrow = x + (size_t)(row_base + m) * IN_CH;

    for (int k0 = 0; k0 < IN_CH; k0 += 4) {
        const int ka = k0 + 2 * half;
        v2f a;
        a.x = xrow[ka];
        a.y = xrow[ka + 1];
        const float* wp = W + (size_t)ka * OUT_CH + m;
#pragma unroll
        for (int t = 0; t < 4; ++t) {
            v2f b;
            b.x = wp[t * 16];
            b.y = wp[OUT_CH + t * 16];
            acc[t] = __builtin_amdgcn_wmma_f32_16x16x4_f32(
                false, a, false, b, (short)0, acc[t], false, false);
        }
    }

    // Store h tile.
#pragma unroll
    for (int t = 0; t < 4; ++t) {
#pragma unroll
        for (int v = 0; v < 8; ++v) {
            const int r = row_base + v + 8 * half;
            h[(size_t)r * OUT_CH + t * 16 + m] = acc[t][v];
        }
    }

    // Fused attention projections: a_src[r] = h[r,:] . att_src  (same for dst).
    // Each (v, half) pair is one row; reduce across the 16 lanes of that half.
#pragma unroll
    for (int v = 0; v < 8; ++v) {
        float ps = 0.f, pd = 0.f;
#pragma unroll
        for (int t = 0; t < 4; ++t) {
            const float hv = acc[t][v];
            ps += hv * att_src[t * 16 + m];
            pd += hv * att_dst[t * 16 + m];
        }
#pragma unroll
        for (int off = 8; off > 0; off >>= 1) {  // masks <16 keep halves separate
            ps += __shfl_xor(ps, off, 32);
            pd += __shfl_xor(pd, off, 32);
        }
        if (m == 0) {
            const int r = row_base + v + 8 * half;
            asrc[r] = ps;
            adst[r] = pd;
        }
    }
}

// ---------------------------------------------------------------------------
// Monotonic unsigned encoding of float for atomicMax-based segment max.
// ---------------------------------------------------------------------------
__device__ __forceinline__ unsigned enc_f(float f) {
    unsigned u = __float_as_uint(f);
    return (u & 0x80000000u) ? ~u : (u | 0x80000000u);
}
__device__ __forceinline__ float dec_f(unsigned u) {
    unsigned b = (u & 0x80000000u) ? (u & 0x7FFFFFFFu) : ~u;
    return __uint_as_float(b);
}

__device__ __forceinline__ void edge_endpoints(const int* __restrict__ ei,
                                               int i, int E, int& s, int& d) {
    if (i < E) { s = ei[i]; d = ei[E + i]; }
    else       { s = i - E; d = i - E; }       // self-loops appended after edges
}

// Pass 1: logits + segment max over dst.
__global__ __launch_bounds__(256) void gat_edge_max(
    const int* __restrict__ ei, const float* __restrict__ asrc,
    const float* __restrict__ adst, float* __restrict__ ew,
    unsigned* __restrict__ menc, int E, int total)
{
    int i = blockIdx.x * 256 + threadIdx.x;
    if (i >= total) return;
    int s, d;
    edge_endpoints(ei, i, E, s, d);
    float t = asrc[s] + adst[d];
    t = (t > 0.f) ? t : NEG_SLOPE * t;
    ew[i] = t;
    atomicMax(&menc[d], enc_f(t));
}

// Pass 2: e = exp(logit - max); denom[dst] += e.
__global__ __launch_bounds__(256) void gat_edge_denom(
    const int* __restrict__ ei, float* __restrict__ ew,
    const unsigned* __restrict__ menc, float* __restrict__ denom,
    int E, int total)
{
    int i = blockIdx.x * 256 + threadIdx.x;
    if (i >= total) return;
    int s, d;
    edge_endpoints(ei, i, E, s, d);
    float w = __expf(ew[i] - dec_f(menc[d]));
    ew[i] = w;
    atomicAdd(&denom[d], w);
}

// Pass 3: out[dst,:] += (e/denom[dst]) * h[src,:]; one thread per (edge, chan).
__global__ __launch_bounds__(256) void gat_edge_aggregate(
    const int* __restrict__ ei, const float* __restrict__ ew,
    const float* __restrict__ denom, const float* __restrict__ h,
    float* __restrict__ outb, int E, int total)
{
    unsigned gid = blockIdx.x * 256u + threadIdx.x;
    int e = (int)(gid >> 6);
    if (e >= total) return;
    int c = (int)(gid & 63u);
    int s, d;
    edge_endpoints(ei, e, E, s, d);
    float w = ew[e] / denom[d];
    atomicAdd(&outb[(size_t)d * OUT_CH + c], w * h[(size_t)s * OUT_CH + c]);
}

// ---------------------------------------------------------------------------
// Pool: LDS partial sums per block (batch is sorted -> low contention), then
// one global flush per block. bias is folded in here.
// ---------------------------------------------------------------------------
__global__ __launch_bounds__(256) void gat_pool(
    const float* __restrict__ outb, const int* __restrict__ batch,
    const float* __restrict__ bias, float* __restrict__ sums,
    float* __restrict__ counts, int n)
{
    __shared__ float ls[N_GRAPHS * OUT_CH];
    __shared__ float lc[N_GRAPHS];
    for (int i = threadIdx.x; i < N_GRAPHS * OUT_CH; i += 256) ls[i] = 0.f;
    if (threadIdx.x < N_GRAPHS) lc[threadIdx.x] = 0.f;
    __syncthreads();

    const int start = blockIdx.x * POOL_CHUNK;
    for (int idx = threadIdx.x; idx < POOL_CHUNK * OUT_CH; idx += 256) {
        int n0 = start + (idx >> 6);
        if (n0 < n) {
            int c = idx & 63;
            int g = batch[n0];
            atomicAdd(&ls[g * OUT_CH + c], outb[(size_t)n0 * OUT_CH + c] + bias[c]);
        }
    }
    for (int idx = threadIdx.x; idx < POOL_CHUNK; idx += 256) {
        int n0 = start + idx;
        if (n0 < n) atomicAdd(&lc[batch[n0]], 1.0f);
    }
    __syncthreads();

    for (int i = threadIdx.x; i < N_GRAPHS * OUT_CH; i += 256)
        atomicAdd(&sums[i], ls[i]);
    if (threadIdx.x < N_GRAPHS) atomicAdd(&counts[threadIdx.x], lc[threadIdx.x]);
}

// Head: logits[g] = (sums[g,:]/count[g]) . W2 + b2
__global__ void gat_head(const float* __restrict__ sums,
                         const float* __restrict__ counts,
                         const float* __restrict__ W2,
                         const float* __restrict__ b2,
                         float* __restrict__ out)
{
    int g = threadIdx.x;
    if (g >= N_GRAPHS) return;
    float cnt = fmaxf(counts[g], 1.0f);
    float acc = b2[0];
    for (int c = 0; c < OUT_CH; ++c)
        acc += (sums[g * OUT_CH + c] / cnt) * W2[c];
    out[g] = acc;
}

// ---------------------------------------------------------------------------
extern "C" void kernel_launch(void* const* d_in, const int* in_sizes, int n_in,
                              void* d_out, int out_size, void* d_ws, size_t ws_size,
                              hipStream_t stream)
{
    const float* x       = (const float*)d_in[0];
    const int*   ei      = (const int*)d_in[1];
    const int*   batch   = (const int*)d_in[2];
    const float* W       = (const float*)d_in[3];
    const float* att_src = (const float*)d_in[4];
    const float* att_dst = (const float*)d_in[5];
    const float* bias    = (const float*)d_in[6];
    const float* W2      = (const float*)d_in[7];
    const float* b2      = (const float*)d_in[8];
    float* out = (float*)d_out;

    const int N = in_sizes[0] / IN_CH;   // 100000
    const int E = in_sizes[1] / 2;       // 1600000
    const int T = E + N;                 // edges + self-loops

    char* ws = (char*)d_ws;
    size_t off = 0;
    auto carve = [&](size_t bytes) -> void* {
        void* p = ws + off;
        off += (bytes + 255) & ~(size_t)255;
        return p;
    };
    float*    h     = (float*)   carve((size_t)N * OUT_CH * 4);
    float*    outb  = (float*)   carve((size_t)N * OUT_CH * 4);
    float*    asrc  = (float*)   carve((size_t)N * 4);
    float*    adst  = (float*)   carve((size_t)N * 4);
    unsigned* menc  = (unsigned*)carve((size_t)N * 4);
    float*    denom = (float*)   carve((size_t)N * 4);
    float*    ew    = (float*)   carve((size_t)T * 4);
    float*    sums  = (float*)   carve((size_t)N_GRAPHS * OUT_CH * 4);
    float*    cnts  = (float*)   carve((size_t)N_GRAPHS * 4);

    // Zero accumulators every call (graph-capture-safe).
    hipMemsetAsync(outb,  0, (size_t)N * OUT_CH * 4, stream);
    hipMemsetAsync(menc,  0, (size_t)N * 4, stream);
    hipMemsetAsync(denom, 0, (size_t)N * 4, stream);
    hipMemsetAsync(sums,  0, (size_t)N_GRAPHS * OUT_CH * 4, stream);
    hipMemsetAsync(cnts,  0, (size_t)N_GRAPHS * 4, stream);

    // 1) GEMM + attention projections
    const int row_tiles = (N + 15) / 16;
    const int gemm_blocks = (row_tiles + 3) / 4;
    gat_gemm_wmma<<<gemm_blocks, 128, 0, stream>>>(x, W, att_src, att_dst,
                                                   h, asrc, adst, N);

    // 2) edge softmax
    const int eb = (T + 255) / 256;
    gat_edge_max<<<eb, 256, 0, stream>>>(ei, asrc, adst, ew, menc, E, T);
    gat_edge_denom<<<eb, 256, 0, stream>>>(ei, ew, menc, denom, E, T);

    const size_t agg_threads = (size_t)T * OUT_CH;
    const int ab = (int)((agg_threads + 255) / 256);
    gat_edge_aggregate<<<ab, 256, 0, stream>>>(ei, ew, denom, h, outb, E, T);

    // 3) pool + head
    const int pb = (N + POOL_CHUNK - 1) / POOL_CHUNK;
    gat_pool<<<pb, 256, 0, stream>>>(outb, batch, bias, sums, cnts, N);
    gat_head<<<1, 64, 0, stream>>>(sums, cnts, W2, b2, out);
}